// PathDev_35227321761793
// MI455X (gfx1250) — compile-verified
//
#include <hip/hip_runtime.h>

typedef float v8f __attribute__((ext_vector_type(8)));
typedef float v4f __attribute__((ext_vector_type(4)));
typedef float v2f __attribute__((ext_vector_type(2)));

#define BATCH  64
#define TLEN   4096
#define DIM    16
#define NCHUNK 64
#define CLEN   (TLEN / NCHUNK)
#define WAVES  8
#define CSTRIDE 20                 // LDS column stride (floats): 16B-aligned columns,
                                   // 20*col mod 64 covers 16 distinct 4-bank groups
#define MATLDS (DIM * CSTRIDE)     // one staged 16x16 matrix in LDS (column-major)

// ---- fragment helpers: a wave holds one 16x16 f32 matrix in WMMA C/D layout ----
// C/D layout: VGPR i: lanes 0-15 -> row i, lanes 16-31 -> row i+8; col = lane & 15.
// A layout (16x4 per K-chunk): lane m (& 15) holds A[m][k0], A[m][k0+1], k0 = 4*kc + 2*hi.
// B layout (4x16 per K-chunk): lane n (& 15) holds B[k0][n], B[k0+1][n].
// LDS staging is COLUMN-major: addr(row, col) = col*CSTRIDE + row, so
//   put_frag   -> 2x ds_store_b128 (one column slice per lane, contiguous)
//   bfrag load -> 4x ds_load_b64   (adjacent rows, same column, contiguous)

__device__ __forceinline__ void put_frag_lds(float* p, v8f m, int lane) {
  const int col = lane & 15, hi = lane >> 4;
  float* q = p + col * CSTRIDE + 8 * hi;   // 16B aligned: col*80 + 32*hi bytes
  v4f lo, hi4;
#pragma unroll
  for (int i = 0; i < 4; ++i) { lo[i] = m[i]; hi4[i] = m[i + 4]; }
  *(v4f*)q = lo;
  *(v4f*)(q + 4) = hi4;
}

__device__ __forceinline__ v8f load_frag_g(const float* p, int lane) {
  const int col = lane & 15, hi = lane >> 4;
  v8f m;
#pragma unroll
  for (int i = 0; i < 8; ++i) m[i] = p[(i + 8 * hi) * DIM + col];
  return m;
}

__device__ __forceinline__ void store_frag_g(float* p, v8f m, int lane) {
  const int col = lane & 15, hi = lane >> 4;
#pragma unroll
  for (int i = 0; i < 8; ++i) p[(i + 8 * hi) * DIM + col] = m[i];
}

// store transpose (orthogonal inverse)
__device__ __forceinline__ void store_frag_g_T(float* p, v8f m, int lane) {
  const int col = lane & 15, hi = lane >> 4;
#pragma unroll
  for (int i = 0; i < 8; ++i) p[col * DIM + (i + 8 * hi)] = m[i];
}

__device__ __forceinline__ v8f ident_frag(int lane) {
  const int col = lane & 15, hi = lane >> 4;
  v8f m;
#pragma unroll
  for (int i = 0; i < 8; ++i) m[i] = ((i + 8 * hi) == col) ? 1.0f : 0.0f;
  return m;
}

// A-fragments (all 4 K-chunks) straight from a row-major global matrix
__device__ __forceinline__ void load_afrag_g(v2f* af, const float* p, int lane) {
  const int m = lane & 15, hi = lane >> 4;
#pragma unroll
  for (int kc = 0; kc < 4; ++kc) {
    const int k0 = kc * 4 + 2 * hi;
    af[kc].x = p[m * DIM + k0];
    af[kc].y = p[m * DIM + k0 + 1];
  }
}

// Batch-load all four fragments into DISTINCT registers so DS loads issue ahead
// of the WMMA chain (progressive dscnt waits, no WAR serialization).
__device__ __forceinline__ void gather_afrags(v2f* a, const float* lA, int lane) {
  const int mn = lane & 15, hi = lane >> 4;
#pragma unroll
  for (int kc = 0; kc < 4; ++kc) {
    const int k0 = kc * 4 + 2 * hi;
    a[kc].x = lA[k0 * CSTRIDE + mn];        // column-major: cols k0, k0+1 of row mn
    a[kc].y = lA[(k0 + 1) * CSTRIDE + mn];
  }
}

__device__ __forceinline__ void gather_bfrags(v2f* b, const float* lB, int lane) {
  const int mn = lane & 15, hi = lane >> 4;
#pragma unroll
  for (int kc = 0; kc < 4; ++kc) {
    const int k0 = kc * 4 + 2 * hi;         // even -> 8B aligned
    b[kc] = *(const v2f*)(lB + mn * CSTRIDE + k0);  // rows k0,k0+1: contiguous b64
  }
}

__device__ __forceinline__ v8f wmma_chain(const v2f* a, const v2f* b) {
  v8f acc = {};
#pragma unroll
  for (int kc = 0; kc < 4; ++kc)
    acc = __builtin_amdgcn_wmma_f32_16x16x4_f32(false, a[kc], false, b[kc], (short)0,
                                                acc, false, false);
  return acc;
}

// D = A @ B, both operands staged in LDS (may be the SAME buffer, e.g. squaring)
__device__ __forceinline__ v8f wave_mm_ab(const float* lA, const float* lB, int lane) {
  v2f a[4], b[4];
  gather_afrags(a, lA, lane);
  gather_bfrags(b, lB, lane);
  return wmma_chain(a, b);
}

// D = A @ B with A-fragments already in registers (no LDS for A)
__device__ __forceinline__ v8f wave_mm_rb(const v2f* af, const float* lB, int lane) {
  v2f b[4];
  gather_bfrags(b, lB, lane);
  return wmma_chain(af, b);
}

// ---------------- Phase 1: per-(batch,chunk) expm + local prefix products -------------
__global__ void __launch_bounds__(WAVES * 32)
phase1_kernel(const float* __restrict__ Z, const float* __restrict__ A1,
              const float* __restrict__ A2, float* __restrict__ outA,
              float* __restrict__ wsP) {
  __shared__ float lds[WAVES * 2 * MATLDS];
  const int lane = threadIdx.x & 31;
  const int wave = threadIdx.x >> 5;
  const int gw   = blockIdx.x * WAVES + wave;
  const int b    = gw / NCHUNK;
  const int c    = gw % NCHUNK;
  float* lA = &lds[wave * 2 * MATLDS];
  float* lB = lA + MATLDS;

  const int col = lane & 15, hi = lane >> 4, mrow = col;
  // skew-symmetric S = A - A^T : keep BOTH C-layout frags (for the Taylor constant
  // term) and A-layout frags (register A-operand of every Horner matmul).
  v8f s1, s2;
#pragma unroll
  for (int i = 0; i < 8; ++i) {
    const int row = i + 8 * hi;
    s1[i] = A1[row * DIM + col] - A1[col * DIM + row];
    s2[i] = A2[row * DIM + col] - A2[col * DIM + row];
  }
  v2f s1a[4], s2a[4];
#pragma unroll
  for (int kc = 0; kc < 4; ++kc) {
    const int k0 = kc * 4 + 2 * hi;
    s1a[kc].x = A1[mrow * DIM + k0] - A1[k0 * DIM + mrow];
    s1a[kc].y = A1[mrow * DIM + k0 + 1] - A1[(k0 + 1) * DIM + mrow];
    s2a[kc].x = A2[mrow * DIM + k0] - A2[k0 * DIM + mrow];
    s2a[kc].y = A2[mrow * DIM + k0 + 1] - A2[(k0 + 1) * DIM + mrow];
  }
  const v8f id = ident_frag(lane);
  v8f R = id;  // running local prefix

  const int t1 = c * CLEN + CLEN;
  int t = c * CLEN;
  if (t == 0) {  // A[b,0] = I (M[:,0] unused); no WMMA in this divergent region
    store_frag_g(outA + (size_t)b * TLEN * (DIM * DIM), R, lane);
    t = 1;
  }
  const float sc = 1.0f / 256.0f;  // scaling-and-squaring: s = 8
  for (; t < t1; ++t) {
    const float z1 = Z[((size_t)b * TLEN + t) * 2 + 0];
    const float z2 = Z[((size_t)b * TLEN + t) * 2 + 1];
    v8f X;
#pragma unroll
    for (int i = 0; i < 8; ++i) X[i] = sc * (z1 * s1[i] + z2 * s2[i]);
    v2f Xa[4];  // A-layout X fragments, pure register math
#pragma unroll
    for (int kc = 0; kc < 4; ++kc) {
      Xa[kc].x = sc * (z1 * s1a[kc].x + z2 * s2a[kc].x);
      Xa[kc].y = sc * (z1 * s1a[kc].y + z2 * s2a[kc].y);
    }

    // Taylor order 6 via Horner: P = I + X(I + X/2(I + ... (I + X/6)))
    v8f P;
#pragma unroll
    for (int i = 0; i < 8; ++i) P[i] = id[i] + X[i] * (1.0f / 6.0f);
#pragma unroll
    for (int k = 5; k >= 1; --k) {
      put_frag_lds(lB, P, lane);
      v8f Q = wave_mm_rb(Xa, lB, lane);  // A operand (X) lives in registers
      const float inv = 1.0f / (float)k;
#pragma unroll
      for (int i = 0; i < 8; ++i) P[i] = id[i] + Q[i] * inv;
    }
    // 8 squarings: M = P^(2^8); single staged copy feeds both operand layouts
#pragma unroll
    for (int s = 0; s < 8; ++s) {
      put_frag_lds(lA, P, lane);
      P = wave_mm_ab(lA, lA, lane);
    }
    // chain: R = R @ M
    put_frag_lds(lA, R, lane);
    put_frag_lds(lB, P, lane);
    R = wave_mm_ab(lA, lB, lane);
    store_frag_g(outA + ((size_t)b * TLEN + t) * (DIM * DIM), R, lane);
  }
  store_frag_g(wsP + ((size_t)b * NCHUNK + c) * (DIM * DIM), R, lane);  // chunk product
}

// ---------------- Phase 2: per-batch exclusive scan of chunk products ----------------
__global__ void __launch_bounds__(WAVES * 32)
phase2_kernel(const float* __restrict__ wsP, float* __restrict__ wsO) {
  __shared__ float lds[WAVES * 2 * MATLDS];
  const int lane = threadIdx.x & 31;
  const int wave = threadIdx.x >> 5;
  const int b    = blockIdx.x * WAVES + wave;
  float* lA = &lds[wave * 2 * MATLDS];
  float* lB = lA + MATLDS;

  v8f O = ident_frag(lane);
  for (int c = 0; c < NCHUNK; ++c) {
    store_frag_g(wsO + ((size_t)b * NCHUNK + c) * (DIM * DIM), O, lane);
    v8f Pc = load_frag_g(wsP + ((size_t)b * NCHUNK + c) * (DIM * DIM), lane);
    put_frag_lds(lA, O, lane);
    put_frag_lds(lB, Pc, lane);
    O = wave_mm_ab(lA, lB, lane);  // last iteration's result unused; keeps EXEC uniform
  }
}

// ------- Phase 3: A_t = O_c @ localR_t; B_inv = A^T (orthogonal group inverse) -------
__global__ void __launch_bounds__(WAVES * 32)
phase3_kernel(const float* __restrict__ wsO, float* __restrict__ outA,
              float* __restrict__ outB) {
  __shared__ float lds[WAVES * MATLDS];
  const int lane = threadIdx.x & 31;
  const int wave = threadIdx.x >> 5;
  const int gw   = blockIdx.x * WAVES + wave;
  const int b    = gw / NCHUNK;
  const int c    = gw % NCHUNK;
  float* lB = &lds[wave * MATLDS];

  // Chunk offset O is the A operand for the whole chunk: A-fragments straight from
  // global into registers, no LDS staging for A at all in this kernel.
  v2f Oa[4];
  load_afrag_g(Oa, wsO + ((size_t)b * NCHUNK + c) * (DIM * DIM), lane);

  const int t0 = c * CLEN, t1 = t0 + CLEN;
  for (int t = t0; t < t1; ++t) {
    float* pa = outA + ((size_t)b * TLEN + t) * (DIM * DIM);
    v8f R = load_frag_g(pa, lane);
    put_frag_lds(lB, R, lane);
    v8f Af = wave_mm_rb(Oa, lB, lane);
    store_frag_g(pa, Af, lane);
    store_frag_g_T(outB + ((size_t)b * TLEN + t) * (DIM * DIM), Af, lane);
  }
}

extern "C" void kernel_launch(void* const* d_in, const int* in_sizes, int n_in,
                              void* d_out, int out_size, void* d_ws, size_t ws_size,
                              hipStream_t stream) {
  (void)in_sizes; (void)n_in; (void)out_size; (void)ws_size;
  const float* Z  = (const float*)d_in[0];
  const float* A1 = (const float*)d_in[1];
  const float* A2 = (const float*)d_in[2];
  float* outA = (float*)d_out;
  float* outB = outA + (size_t)BATCH * TLEN * DIM * DIM;
  float* wsP  = (float*)d_ws;                                  // [B, NCHUNK, 16,16]
  float* wsO  = wsP + (size_t)BATCH * NCHUNK * DIM * DIM;      // [B, NCHUNK, 16,16]

  dim3 blk(WAVES * 32);
  phase1_kernel<<<(BATCH * NCHUNK) / WAVES, blk, 0, stream>>>(Z, A1, A2, outA, wsP);
  phase2_kernel<<<BATCH / WAVES, blk, 0, stream>>>(wsP, wsO);
  phase3_kernel<<<(BATCH * NCHUNK) / WAVES, blk, 0, stream>>>(wsO, outA, outB);
}